// KarmonicFilterLoss_44547400794372
// MI455X (gfx1250) — compile-verified
//
#include <hip/hip_runtime.h>
#include <math.h>

// ---------------------------------------------------------------------------
// KarmonicFilterLoss for MI455X (gfx1250, wave32).
//
// Heavy part: 6 modes x 8192x8192 Gram (inner dim 8, fp32) + exp epilogue.
// Inputs fit in L2 (1.5 MB) -> compute bound. Gram tiles computed with
// V_WMMA_F32_16X16X4_F32 (2 per 16x16 tile, K=8); epilogue kept minimal:
//   S = T*log2(e);  A pre-scaled by +2S;  norms stored as -S*|x|^2,
//   bias fed through the WMMA C operand  =>  c[v] = -S*d2,
//   p = min(exp2(c), 1)  (== exp2(min(c,0)), avoids canonicalize-before-min).
// K-components permuted {0,1|4,5},{2,3|6,7} so each lane's A/B fragment is a
// single contiguous 16B load (one global_load_b128 per tile).
// Diagonal tile peeled; loop bounds readfirstlane'd so control stays on SALU.
// ---------------------------------------------------------------------------

typedef __attribute__((ext_vector_type(2))) float v2f;
typedef __attribute__((ext_vector_type(4))) float v4f;
typedef __attribute__((ext_vector_type(8))) float v8f;

#define BN     8192
#define MODES  6
#define FDIM   48    // 2*K*M
#define CDIM   8     // 2*K per mode
#define SCALE2 2.8853900817779268f   // S  = T*log2(e), T = 2
#define ASCALE 5.7707801635558537f   // 2S (folded into A operand)
#define LN2    0.6931471805599453f

// ---------------------------------------------------------------------------
// Kernel 1: per-(mode,row) scaled squared norms (-S*|x|^2), mode-major;
// zero the per-mode accumulators.
// ---------------------------------------------------------------------------
__global__ void kfl_precompute(const float* __restrict__ x,
                               float* __restrict__ nsq,
                               float* __restrict__ modeSum) {
  if (blockIdx.x == 0 && threadIdx.x < MODES) modeSum[threadIdx.x] = 0.0f;
  int idx = blockIdx.x * blockDim.x + threadIdx.x;
  if (idx < BN * MODES) {
    int b = idx & (BN - 1);
    int m = idx >> 13;                  // idx / 8192
    const float* p = x + b * FDIM + m * CDIM;
    float s = 0.0f;
#pragma unroll
    for (int c = 0; c < CDIM; ++c) { float v = p[c]; s += v * v; }
    nsq[m * BN + b] = -SCALE2 * s;      // mode-major for coalesced col loads
  }
}

// ---------------------------------------------------------------------------
// Kernel 2: per-mode uniformity. Each wave owns 16 rows of one mode, loops
// over all 512 column tiles; 16x16 Gram tile = 2x v_wmma_f32_16x16x4_f32.
//
// f32 WMMA operand layout (wave32): A(16x4): lane L holds row M=L%16,
// VGPR0/1 = K = (L<16 ? {0,1} : {2,3}). B(4x16): lane L holds col N=L%16,
// same K split. C/D(16x16): VGPR v, lane L: row M = v + 8*(L>=16), col L%16.
// We permute the 8 physical components so WMMA0 consumes {0,1 | 4,5} and
// WMMA1 consumes {2,3 | 6,7}: each lane's fragment pair = contiguous float4
// at byte offset 16*hi (dot product is permutation-invariant when A and B
// use the same assignment).
// ---------------------------------------------------------------------------
__global__ void __launch_bounds__(256)
kfl_unif(const float* __restrict__ x,
         const float* __restrict__ nsq,
         float* __restrict__ modeSum) {
  const int m      = blockIdx.y;
  const int wave   = threadIdx.x >> 5;
  const int lane   = threadIdx.x & 31;
  const int lane15 = lane & 15;
  const int hi     = lane >> 4;
  const int ibase  = blockIdx.x * 128 + wave * 16;
  const float* __restrict__ nsqm = nsq + m * BN;

  // A fragments for this wave's 16 rows (one b128), pre-scaled by 2S.
  const float* pA = x + (ibase + lane15) * FDIM + m * CDIM + hi * 4;
  v4f aa = *(const v4f*)(pA);
  aa *= ASCALE;
  v2f a0 = __builtin_shufflevector(aa, aa, 0, 1);
  v2f a1 = __builtin_shufflevector(aa, aa, 2, 3);

  float sqi[8];
#pragma unroll
  for (int v = 0; v < 8; ++v) sqi[v] = nsqm[ibase + hi * 8 + v];

  float acc[8];
#pragma unroll
  for (int v = 0; v < 8; ++v) acc[v] = 0.0f;

  // One 16x16 tile step.  diag==true only for the single peeled iteration
  // (inlined with constant bool, so the mask code exists once).
  auto tile = [&](int jt, bool diag) {
    const int col   = jt * 16 + lane15;
    const float* pB = x + col * FDIM + m * CDIM + hi * 4;
    v4f bb = *(const v4f*)(pB);                    // one global_load_b128
    v2f b0 = __builtin_shufflevector(bb, bb, 0, 1);
    v2f b1 = __builtin_shufflevector(bb, bb, 2, 3);
    float sqj = nsqm[col];

    // Bias through the WMMA accumulator: C = -S(|xi|^2+|xj|^2),
    // so after two WMMAs c[v] = -S*d2 (exact up to fp32 rounding).
    v8f c;
#pragma unroll
    for (int v = 0; v < 8; ++v) c[v] = sqi[v] + sqj;
    c = __builtin_amdgcn_wmma_f32_16x16x4_f32(false, a0, false, b0,
                                              (short)0, c, false, false);
    c = __builtin_amdgcn_wmma_f32_16x16x4_f32(false, a1, false, b1,
                                              (short)0, c, false, false);
#pragma unroll
    for (int v = 0; v < 8; ++v) {
      // exp2(min(c,0)) == min(exp2(c),1); exp output is canonical, so the
      // min needs no canonicalize op.
      float p = fminf(__builtin_amdgcn_exp2f(c[v]), 1.0f);
      if (diag && lane15 == (hi * 8 + v)) p = 0.0f;   // exclude i==j
      acc[v] += p;
    }
  };

  // Wave-uniform bounds (readfirstlane -> SGPR) keep loop control on SALU.
  const int diagJt = __builtin_amdgcn_readfirstlane(ibase >> 4);
#pragma unroll 2
  for (int jt = 0; jt < diagJt; ++jt) tile(jt, false);
  tile(diagJt, true);
#pragma unroll 2
  for (int jt = diagJt + 1; jt < BN / 16; ++jt) tile(jt, false);

  // Sum over the 16 columns held per half-wave; halves are independent under
  // xor masks 1/2/4/8, so lanes 0 and 16 end with rows v and v+8 totals.
  float lsum = 0.0f;
#pragma unroll
  for (int v = 0; v < 8; ++v) {
    float s = acc[v];
    s += __shfl_xor(s, 1);
    s += __shfl_xor(s, 2);
    s += __shfl_xor(s, 4);
    s += __shfl_xor(s, 8);
    if (lane15 == 0) lsum += __builtin_amdgcn_logf(s) * LN2;  // ln via log2
  }
  if (lane15 == 0) atomicAdd(&modeSum[m], lsum);
}

// ---------------------------------------------------------------------------
// Kernel 3: circular-spread term + final combine (single block).
// ---------------------------------------------------------------------------
__global__ void kfl_finalize(const float* __restrict__ angles,
                             const float* __restrict__ modeSum,
                             float* __restrict__ out) {
  __shared__ float ssum[8];    // [0..3] sum sin, [4..7] sum cos
  __shared__ float scov[10];   // upper-tri incl. diag of 4x4 s^T s
  __shared__ float mu[4];
  const int t = threadIdx.x;
  if (t < 8)  ssum[t] = 0.0f;
  if (t < 10) scov[t] = 0.0f;
  __syncthreads();

  float ps[4] = {0, 0, 0, 0}, pc[4] = {0, 0, 0, 0};
  for (int b = t; b < BN; b += blockDim.x) {
#pragma unroll
    for (int k = 0; k < 4; ++k) {
      float a = angles[b * 4 + k];
      ps[k] += __sinf(a);
      pc[k] += __cosf(a);
    }
  }
#pragma unroll
  for (int k = 0; k < 4; ++k) {
    atomicAdd(&ssum[k], ps[k]);
    atomicAdd(&ssum[4 + k], pc[k]);
  }
  __syncthreads();
  if (t < 4) mu[t] = atan2f(ssum[t], ssum[4 + t]);  // 1/B scale cancels
  __syncthreads();

  float pp[10];
#pragma unroll
  for (int i = 0; i < 10; ++i) pp[i] = 0.0f;
  for (int b = t; b < BN; b += blockDim.x) {
    float s[4];
#pragma unroll
    for (int k = 0; k < 4; ++k) s[k] = __sinf(angles[b * 4 + k] - mu[k]);
    int idx = 0;
#pragma unroll
    for (int i = 0; i < 4; ++i)
#pragma unroll
      for (int j = i; j < 4; ++j) pp[idx++] += s[i] * s[j];
  }
#pragma unroll
  for (int i = 0; i < 10; ++i) atomicAdd(&scov[i], pp[i]);
  __syncthreads();

  if (t == 0) {
    float cov[4][4];
    int idx = 0;
    for (int i = 0; i < 4; ++i)
      for (int j = i; j < 4; ++j) {
        float c = scov[idx++] / (float)BN;
        cov[i][j] = c;
        cov[j][i] = c;
      }
    float spread = 0.0f;
    for (int i = 0; i < 4; ++i)
      for (int j = i + 1; j < 4; ++j) {
        float den = sqrtf(cov[i][i] * cov[j][j] + 1e-8f);
        float cr  = cov[i][j] / den;
        spread += cr * cr;
      }
    spread *= (1.0f / 6.0f);

    // Karmonic weights: lam_n = 2 - 2 cos(2*pi*(n+1)/12), n = 0..5
    float lam[MODES];
    for (int n = 0; n < MODES; ++n)
      lam[n] = 2.0f - 2.0f * __cosf(2.0f * 3.14159265358979f * (n + 1) / 12.0f);
    float lmax = lam[0];
    for (int n = 1; n < MODES; ++n) lmax = fmaxf(lmax, lam[n]);

    const float logB1 = __logf((float)(BN - 1));
    float totalU = 0.0f;
    for (int n = 0; n < MODES; ++n) {
      float un = modeSum[n] / (float)BN - logB1;   // mean lse - log(B-1)
      out[3 + n] = un;
      float w = (lam[n] - lam[0]) / (lmax - lam[0]);
      totalU += w * un;
    }
    out[0] = totalU;
    out[1] = spread;
    out[2] = totalU + spread;   // SPREAD_W = 1
  }
}

// ---------------------------------------------------------------------------
extern "C" void kernel_launch(void* const* d_in, const int* in_sizes, int n_in,
                              void* d_out, int out_size, void* d_ws, size_t ws_size,
                              hipStream_t stream) {
  const float* angles = (const float*)d_in[0];   // (8192, 4)  fp32
  const float* fe     = (const float*)d_in[1];   // (8192, 48) fp32
  float* out = (float*)d_out;                    // 9 fp32 outputs

  float* nsq     = (float*)d_ws;                 // MODES*BN floats (-S*|x|^2)
  float* modeSum = nsq + MODES * BN;             // MODES floats

  (void)in_sizes; (void)n_in; (void)out_size; (void)ws_size;

  kfl_precompute<<<(BN * MODES) / 256, 256, 0, stream>>>(fe, nsq, modeSum);

  dim3 gridU(BN / 128, MODES);                   // 64 row-blocks x 6 modes
  kfl_unif<<<gridU, 256, 0, stream>>>(fe, nsq, modeSum);

  kfl_finalize<<<1, 256, 0, stream>>>(angles, modeSum, out);
}